// TaskSpecificGCN_1039382085841
// MI455X (gfx1250) — compile-verified
//
#include <hip/hip_runtime.h>
#include <stdint.h>

// ---------------------------------------------------------------------------
// Vector types
// ---------------------------------------------------------------------------
typedef float v2f __attribute__((ext_vector_type(2)));
typedef float v8f __attribute__((ext_vector_type(8)));
typedef unsigned int u32x4 __attribute__((ext_vector_type(4)));
typedef int i32x4 __attribute__((ext_vector_type(4)));
typedef int i32x8 __attribute__((ext_vector_type(8)));

__device__ __forceinline__ void atomic_add_f32(float* p, float v) {
    // Relaxed, agent-scope FP32 atomic -> global_atomic_add_f32 on gfx1250.
    __hip_atomic_fetch_add(p, v, __ATOMIC_RELAXED, __HIP_MEMORY_SCOPE_AGENT);
}

// ---------------------------------------------------------------------------
// Zero fill
// ---------------------------------------------------------------------------
__global__ void k_zero(float* __restrict__ p, int n) {
    int i = blockIdx.x * blockDim.x + threadIdx.x;
    if (i < n) p[i] = 0.0f;
}

// ---------------------------------------------------------------------------
// Degree accumulation over edge targets (in-degree), then dis = rsqrt(deg+1)
// ---------------------------------------------------------------------------
__global__ void k_deg(const long long* __restrict__ ei, float* __restrict__ deg,
                      int n_edges) {
    int e = blockIdx.x * blockDim.x + threadIdx.x;
    if (e < n_edges) {
        int d = (int)ei[(size_t)n_edges + e];   // dst row
        atomic_add_f32(&deg[d], 1.0f);
    }
}

__global__ void k_dis(float* __restrict__ deg_dis, int n) {
    int i = blockIdx.x * blockDim.x + threadIdx.x;
    if (i < n) {
        float d = deg_dis[i] + 1.0f;            // + self loop
        deg_dis[i] = rsqrtf(d);
    }
}

// ---------------------------------------------------------------------------
// TDM: DMA W [rows x 64 f32, row-major, contiguous] into LDS with a 4-DWORD
// pad every 64 DWORDs (68-float LDS row stride -> bank-conflict-free B reads).
// D# layout per CDNA5 ISA §8.3/8.4. 2-D tensor: higher groups zero.
// This toolchain's builtin takes 6 args (g0, g1, g2, g3, g4, cpol).
// ---------------------------------------------------------------------------
#if __has_builtin(__builtin_amdgcn_tensor_load_to_lds)
#define USE_TDM 1
__device__ __forceinline__ void tdm_load_w(const float* W, uint32_t lds_off,
                                           int rows) {
    uint64_t ga = (uint64_t)(uintptr_t)W;
    u32x4 g0;
    g0[0] = 1u;                                    // count=1 (valid user D#)
    g0[1] = lds_off;                               // lds_addr (bytes)
    g0[2] = (uint32_t)ga;                          // global_addr[31:0]
    g0[3] = (uint32_t)((ga >> 32) & 0x01FFFFFFu)   // global_addr[56:32]
          | (2u << 30);                            // type = 2 ("image")
    i32x8 g1;
    g1[0] = (int)((2u << 16)                       // data_size = 4 bytes
          |       (1u << 20)                       // pad_enable
          |       (5u << 22)                       // pad_interval: 64 DWORDs
          |       (3u << 25));                     // pad_amount: 4 DWORDs
    g1[1] = (int)((64u & 0xFFFFu) << 16);          // tensor_dim0 = 64 (lo16)
    g1[2] = (int)(((uint32_t)rows & 0xFFFFu) << 16); // dim0 hi=0 | tensor_dim1 lo16
    g1[3] = (int)(64u << 16);                      // dim1 hi=0 | tile_dim0 = 64
    g1[4] = (int)((uint32_t)rows & 0xFFFFu);       // tile_dim1 = rows, tile_dim2=0
    g1[5] = 64;                                    // tensor_dim0_stride lo32
    g1[6] = 0;                                     // stride hi | dim1_stride lo16
    g1[7] = 0;                                     // tensor_dim1_stride hi32
    i32x4 g2 = {0, 0, 0, 0};
    i32x4 g3 = {0, 0, 0, 0};
    i32x8 g4 = {0, 0, 0, 0, 0, 0, 0, 0};
    __builtin_amdgcn_tensor_load_to_lds(g0, g1, g2, g3, g4, 0);
}
#else
#define USE_TDM 0
#endif

// ---------------------------------------------------------------------------
// H = X @ W^T, X:[n,64], W:[OUT,64] row-major.  One wave -> one 16x16 output
// tile; K=64 as 16 chained v_wmma_f32_16x16x4_f32.  W staged in LDS by TDM.
//
// A frag (16x4 MxK): lane&15 = M, K-pair = ks*4 + 2*(lane>>4) -> float2 load.
// B frag (4x16 KxN): lane&15 = N, same K striping -> float2 LDS load.
// C/D: VGPR r -> M = r + 8*(lane>>4), N = lane&15.
// ---------------------------------------------------------------------------
template <int OUT>
__global__ void k_gemm_xwt(const float* __restrict__ X,
                           const float* __restrict__ W,
                           float* __restrict__ H, int n_nodes) {
    constexpr int WS = 68;                         // padded LDS row stride
    __shared__ float wlds[OUT * WS];

#if USE_TDM
    if (threadIdx.x < 32) {                        // one wave issues the DMA
        tdm_load_w(W, (uint32_t)(uintptr_t)&wlds[0], OUT);
        __builtin_amdgcn_s_wait_tensorcnt(0);
    }
#else
    for (int i = threadIdx.x; i < OUT * 64; i += blockDim.x)
        wlds[(i >> 6) * WS + (i & 63)] = W[i];
#endif
    __syncthreads();

    const int lane = threadIdx.x & 31;
    const int wave = threadIdx.x >> 5;             // output-column tile
    const int half = lane >> 4;
    const int m    = lane & 15;
    int row = blockIdx.x * 16 + m;
    if (row >= n_nodes) row = n_nodes - 1;         // clamp, keep EXEC all-ones

    const float* __restrict__ xrow = X + (size_t)row * 64;
    const float* __restrict__ wrow = &wlds[(wave * 16 + m) * WS];

    v8f acc = {};
#pragma unroll
    for (int ks = 0; ks < 16; ++ks) {
        const int k = ks * 4 + 2 * half;
        v2f a, b;
        a.x = xrow[k]; a.y = xrow[k + 1];
        b.x = wrow[k]; b.y = wrow[k + 1];
        acc = __builtin_amdgcn_wmma_f32_16x16x4_f32(
            false, a, false, b, (short)0, acc, false, false);
    }

#pragma unroll
    for (int r = 0; r < 8; ++r) {
        const int mrow = blockIdx.x * 16 + r + 8 * half;
        if (mrow < n_nodes)
            H[(size_t)mrow * OUT + wave * 16 + m] = acc[r];
    }
}

// ---------------------------------------------------------------------------
// Edge scatter:  agg[dst] += h[src] * dis[src]*dis[dst]
// F/4 threads per edge -> coalesced 4*F-byte row gathers, HW FP32 atomics out.
// ---------------------------------------------------------------------------
template <int F>
__global__ void k_scatter(const long long* __restrict__ ei,
                          const float* __restrict__ h,
                          const float* __restrict__ dis,
                          float* __restrict__ agg, int n_edges) {
    constexpr int TPE = F / 4;
    const int idx = blockIdx.x * blockDim.x + threadIdx.x;
    if (idx >= n_edges * TPE) return;
    const int e = idx / TPE;
    const int q = idx - e * TPE;

    const int s = (int)ei[e];
    const int d = (int)ei[(size_t)n_edges + e];
    const float w = dis[s] * dis[d];

    const float4 v = *(const float4*)(h + (size_t)s * F + q * 4);
    float* o = agg + (size_t)d * F + q * 4;
    atomic_add_f32(o + 0, v.x * w);
    atomic_add_f32(o + 1, v.y * w);
    atomic_add_f32(o + 2, v.z * w);
    atomic_add_f32(o + 3, v.w * w);
}

// ---------------------------------------------------------------------------
// Epilogue: out = agg + h*dis^2 (self loop) + bias, optional ReLU / dropout.
// ---------------------------------------------------------------------------
__device__ __forceinline__ uint32_t hash_u32(uint32_t x) {
    x *= 2654435761u; x ^= x >> 16;
    x *= 2246822519u; x ^= x >> 13;
    x *= 3266489917u; x ^= x >> 16;
    return x;
}

template <int F, bool RELU, bool DROP>
__global__ void k_epilogue(const float* __restrict__ agg,
                           const float* __restrict__ h,
                           const float* __restrict__ dis,
                           const float* __restrict__ bias,
                           float* __restrict__ out, int n_nodes) {
    const int idx = blockIdx.x * blockDim.x + threadIdx.x;
    if (idx >= n_nodes * F) return;
    const int node = idx / F;
    const int f    = idx - node * F;
    const float di = dis[node];
    float v = agg[idx] + h[idx] * di * di + bias[f];
    if (RELU) v = fmaxf(v, 0.0f);
    if (DROP) {
        const uint32_t r = hash_u32((uint32_t)idx) & 0x00FFFFFFu;
        v = (r < 13421773u) ? v * 1.25f : 0.0f;   // keep-prob 0.8
    }
    out[idx] = v;
}

// ---------------------------------------------------------------------------
// Host-side orchestration
// ---------------------------------------------------------------------------
extern "C" void kernel_launch(void* const* d_in, const int* in_sizes, int n_in,
                              void* d_out, int out_size, void* d_ws, size_t ws_size,
                              hipStream_t stream) {
    const float*     x  = (const float*)d_in[0];       // [N,64]
    const long long* ei = (const long long*)d_in[1];   // [2,E] int64
    const float*     W1 = (const float*)d_in[2];       // [64,64]
    const float*     b1 = (const float*)d_in[3];       // [64]
    const float*     W2 = (const float*)d_in[4];       // [64,64]
    const float*     b2 = (const float*)d_in[5];       // [64]
    const float*     W3 = (const float*)d_in[6];       // [16,64]
    const float*     b3 = (const float*)d_in[7];       // [16]

    const int n_nodes = in_sizes[0] / 64;
    const int n_edges = in_sizes[1] / 2;

    // Workspace layout (floats): dis | hbuf [N*64] | abuf [N*64]
    float* ws   = (float*)d_ws;
    float* dis  = ws;
    float* hbuf = ws + 102400;
    float* abuf = hbuf + (size_t)n_nodes * 64;
    float* out  = (float*)d_out;

    const int T = 256;
    const int nf64 = n_nodes * 64;
    const int nf16 = n_nodes * 16;
    const int g_n    = (n_nodes + T - 1) / T;
    const int g_e    = (n_edges + T - 1) / T;
    const int g_nf64 = (nf64 + T - 1) / T;
    const int g_nf16 = (nf16 + T - 1) / T;
    const int g_sc64 = (n_edges * 16 + T - 1) / T;
    const int g_sc16 = (n_edges * 4  + T - 1) / T;
    const int g_mm   = (n_nodes + 15) / 16;

    // --- degrees -> dis (shared by all layers) ---
    k_zero<<<g_n, T, 0, stream>>>(dis, n_nodes);
    k_deg<<<g_e, T, 0, stream>>>(ei, dis, n_edges);
    k_dis<<<g_n, T, 0, stream>>>(dis, n_nodes);

    // --- layer 1 ---
    k_gemm_xwt<64><<<g_mm, 128, 0, stream>>>(x, W1, hbuf, n_nodes);
    k_zero<<<g_nf64, T, 0, stream>>>(abuf, nf64);
    k_scatter<64><<<g_sc64, T, 0, stream>>>(ei, hbuf, dis, abuf, n_edges);
    k_epilogue<64, true, false><<<g_nf64, T, 0, stream>>>(abuf, hbuf, dis, b1, abuf, n_nodes);

    // --- layer 2 (+ dropout) ---
    k_gemm_xwt<64><<<g_mm, 128, 0, stream>>>(abuf, W2, hbuf, n_nodes);
    k_zero<<<g_nf64, T, 0, stream>>>(abuf, nf64);
    k_scatter<64><<<g_sc64, T, 0, stream>>>(ei, hbuf, dis, abuf, n_edges);
    k_epilogue<64, true, true><<<g_nf64, T, 0, stream>>>(abuf, hbuf, dis, b2, abuf, n_nodes);

    // --- layer 3 ---
    k_gemm_xwt<16><<<g_mm, 32, 0, stream>>>(abuf, W3, hbuf, n_nodes);
    k_zero<<<g_nf16, T, 0, stream>>>(out, nf16);
    k_scatter<16><<<g_sc16, T, 0, stream>>>(ei, hbuf, dis, out, n_edges);
    k_epilogue<16, false, false><<<g_nf16, T, 0, stream>>>(out, hbuf, dis, b3, out, n_nodes);
}